// DecGreenNet_product_CP3_90967407329573
// MI455X (gfx1250) — compile-verified
//
#include <hip/hip_runtime.h>
#include <math.h>

typedef __attribute__((ext_vector_type(2))) float v2f;
typedef __attribute__((ext_vector_type(8))) float v8f;

typedef __attribute__((address_space(1))) int gint_t;   // global (device) int
typedef __attribute__((address_space(3))) int lint_t;   // LDS int

#define NB      65536   // batch
#define HH      512     // hidden / s0^3
#define NQ      8192    // quad nodes
#define HQ      128     // quad hidden
#define PI_F    3.14159265358979323846f
#define INV2PI  0.15915494309189535f

// Feature probes ------------------------------------------------------------
#if defined(__has_builtin)
# if __has_builtin(__builtin_amdgcn_global_load_async_to_lds_b32)
#  define HAS_ASYNC_LDS 1
# endif
#endif
#ifndef HAS_ASYNC_LDS
# define HAS_ASYNC_LDS 0
#endif

__device__ __forceinline__ void wait_async0() {
#if defined(__has_builtin) && __has_builtin(__builtin_amdgcn_s_wait_asynccnt)
    __builtin_amdgcn_s_wait_asynccnt(0);
#else
    asm volatile("s_wait_asynccnt 0x0" ::: "memory");
#endif
}

// Native gfx1250 transcendentals (V_TANH_F32 / V_SIN_F32) with libm fallback.
__device__ __forceinline__ float fast_tanh(float x) {
#if defined(__has_builtin) && __has_builtin(__builtin_amdgcn_tanhf)
    return __builtin_amdgcn_tanhf(x);
#else
    return tanhf(x);
#endif
}
__device__ __forceinline__ float fast_sin(float x) {   // x in radians, small range
#if defined(__has_builtin) && __has_builtin(__builtin_amdgcn_sinf)
    return __builtin_amdgcn_sinf(x * INV2PI);          // HW sin takes revolutions
#else
    return sinf(x);
#endif
}

// XOR-lane reduction step via DS_SWIZZLE_B32 (group-of-32 xor patterns).
#define SWZ_ADD(v, imm)                                                        \
    (v) += __int_as_float(__builtin_amdgcn_ds_swizzle(__float_as_int(v), (imm)))

// workspace layout (floats)
#define WS_T    0       // t[3][128]  : sum_n y_n * tanh(x_n*Wq1+bq1)
#define WS_Y    384     // Y[3]       : sum_n y_n   (+1 pad)
#define WS_V    388     // v[512]     : Wx2 @ rhs
#define WS_C    900     // c          : bx2 . rhs

// ---------------- zero accumulators ----------------
__global__ void kZero(float* __restrict__ ws) {
    int i = threadIdx.x;
    if (i < 388) ws[i] = 0.0f;
}

// ---------------- per-branch node reduction ----------------
// grid: 96 blocks = 3 branches x 32 chunks of 256 nodes; 128 threads
__global__ void kBranch(const float* __restrict__ eq,
                        const float* __restrict__ qx0,
                        const float* __restrict__ qx1,
                        const float* __restrict__ qx2,
                        const float* __restrict__ Wq01, const float* __restrict__ bq01,
                        const float* __restrict__ Wq11, const float* __restrict__ bq11,
                        const float* __restrict__ Wq21, const float* __restrict__ bq21,
                        float* __restrict__ ws) {
    const int branch = blockIdx.x >> 5;
    const int chunk  = blockIdx.x & 31;
    const float* qx = (branch == 0) ? qx0 : (branch == 1) ? qx1 : qx2;
    const float* W1 = (branch == 0) ? Wq01 : (branch == 1) ? Wq11 : Wq21;
    const float* b1 = (branch == 0) ? bq01 : (branch == 1) ? bq11 : bq21;

    __shared__ float xl[256];
    __shared__ float yl[256];
    const int tid = threadIdx.x;
    const float eqv = eq[0];
    for (int i = tid; i < 256; i += 128) {
        float xv = qx[chunk * 256 + i];
        xl[i] = xv;
        yl[i] = fast_sin(PI_F * eqv * xv);
    }
    __syncthreads();

    // thread tid owns hidden unit k = tid
    const float w = W1[tid];
    const float b = b1[tid];
    float tsum = 0.0f;
    #pragma unroll 4
    for (int j = 0; j < 256; ++j)
        tsum += yl[j] * fast_tanh(xl[j] * w + b);
    atomicAdd(&ws[WS_T + branch * 128 + tid], tsum);

    if (tid == 0) {
        float ys = 0.0f;
        for (int j = 0; j < 256; ++j) ys += yl[j];
        atomicAdd(&ws[WS_Y + branch], ys);
    }
}

// ---------------- combine: s -> rhs -> v, c ----------------
// single block, 512 threads
__global__ void kCombine(const float* __restrict__ Wq02, const float* __restrict__ bq02,
                         const float* __restrict__ Wq12, const float* __restrict__ bq12,
                         const float* __restrict__ Wq22, const float* __restrict__ bq22,
                         const float* __restrict__ Wx2,  const float* __restrict__ bx2,
                         float* __restrict__ ws) {
    __shared__ float s_sh[384];    // s_a|s_c|s_e, each [8*16]
    __shared__ float rhs_sh[512];
    const int tid = threadIdx.x;

    if (tid < 384) {
        const int branch = tid >> 7;
        const int j = tid & 127;
        const float* W2 = (branch == 0) ? Wq02 : (branch == 1) ? Wq12 : Wq22;
        const float* b2 = (branch == 0) ? bq02 : (branch == 1) ? bq12 : bq22;
        const float* t  = ws + WS_T + branch * 128;
        const float Yv  = ws[WS_Y + branch];
        float acc = Yv * b2[j];
        for (int k = 0; k < 128; ++k)
            acc += t[k] * W2[k * 128 + j];
        s_sh[tid] = acc;
    }
    __syncthreads();

    {   // rhs[b*64 + d*8 + f] = sum_x s_a[b,x] s_c[d,x] s_e[f,x]
        const int b = tid >> 6, d = (tid >> 3) & 7, f = tid & 7;
        const float* sa = s_sh;
        const float* sc = s_sh + 128;
        const float* se = s_sh + 256;
        float acc = 0.0f;
        #pragma unroll
        for (int x = 0; x < 16; ++x)
            acc += sa[b * 16 + x] * sc[d * 16 + x] * se[f * 16 + x];
        rhs_sh[tid] = acc;
    }
    __syncthreads();

    {   // v[h] = Wx2[h,:] . rhs
        const float* row = Wx2 + tid * 512;
        float acc = 0.0f;
        for (int j = 0; j < 512; ++j) acc += row[j] * rhs_sh[j];
        ws[WS_V + tid] = acc;
    }
    if (tid == 0) {
        float acc = 0.0f;
        for (int j = 0; j < 512; ++j) acc += bx2[j] * rhs_sh[j];
        ws[WS_C] = acc;
    }
}

// ---------------- main: out[b] = sum_h tanh(x W1 + b1)[h] * v[h] + c ----------------
// 1024 blocks x 128 threads; each wave handles a 16-row batch tile via
// V_WMMA_F32_16X16X4_F32 (exact fp32). The K=4 pad slot carries the bias:
// A[*,3] = 1.0 and sWi[col*4+3] = bx1[col], so h1 = x.W + bias comes straight
// out of the WMMA with a loop-invariant zero C operand.
__global__ void kMain(const float* __restrict__ x,     // [B,3]
                      const float* __restrict__ Wx1,   // [3,512]
                      const float* __restrict__ bx1,   // [512]
                      const float* __restrict__ ws,
                      float* __restrict__ out) {
    __shared__ float sWi[4 * 512];   // [col][k] interleaved; k=3 holds bias
    __shared__ float sV[512];
    const int tid = threadIdx.x;

#if HAS_ASYNC_LDS
    // Per-lane async DMA gather->LDS scatter (ASYNCcnt): does the K-interleave
    // transpose (and bias-row placement) in flight.
    for (int e = tid; e < 1536; e += 128) {
        __builtin_amdgcn_global_load_async_to_lds_b32(
            (gint_t*)(Wx1 + e),
            (lint_t*)&sWi[((e & 511) << 2) + (e >> 9)],
            0, 0);
    }
    for (int i = tid; i < 512; i += 128) {
        __builtin_amdgcn_global_load_async_to_lds_b32(
            (gint_t*)(bx1 + i), (lint_t*)&sWi[(i << 2) + 3], 0, 0);
        __builtin_amdgcn_global_load_async_to_lds_b32(
            (gint_t*)(ws + WS_V + i), (lint_t*)&sV[i], 0, 0);
    }
    wait_async0();
#else
    for (int e = tid; e < 1536; e += 128)
        sWi[((e & 511) << 2) + (e >> 9)] = Wx1[e];
    for (int i = tid; i < 512; i += 128) {
        sWi[(i << 2) + 3] = bx1[i];
        sV[i] = ws[WS_V + i];
    }
#endif
    __syncthreads();

    const int lane   = tid & 31;
    const int wave   = tid >> 5;
    const int lane16 = lane & 15;
    const int b0     = blockIdx.x * 64 + wave * 16;
    const int row    = b0 + lane16;
    const int koff   = (lane & 16) >> 3;     // 0 for K0/K1 half, 2 for K2/K3 half

    // A fragment, 16x4 (MxK) f32: VGPR0 = K0|K2, VGPR1 = K1|K3 across lane
    // halves. K3 element = 1.0 so the bias row of B is accumulated.
    v2f a;
    if (lane < 16) { a.x = x[row * 3 + 0]; a.y = x[row * 3 + 1]; }
    else           { a.x = x[row * 3 + 2]; a.y = 1.0f; }

    const float cval = ws[WS_C];
    const v8f czero = {};
    float acc[8];
    #pragma unroll
    for (int r = 0; r < 8; ++r) acc[r] = 0.0f;

    #pragma unroll 2
    for (int ht = 0; ht < 32; ++ht) {
        const int col = ht * 16 + lane16;
        // B fragment (incl. bias row): single aligned 8-byte LDS load
        const v2f bf = *(const v2f*)&sWi[(col << 2) + koff];

        v8f d = __builtin_amdgcn_wmma_f32_16x16x4_f32(
            /*neg_a=*/false, a, /*neg_b=*/false, bf,
            /*c_mod=*/(short)0, czero, /*reuse_a=*/false, /*reuse_b=*/false);

        const float vv = sV[col];
        #pragma unroll
        for (int r = 0; r < 8; ++r)
            acc[r] += fast_tanh(d[r]) * vv;
    }

    // reduce over the 16 columns held by each lane half (D layout: lanes 0-15
    // hold rows 0-7, lanes 16-31 hold rows 8-15). XOR masks <= 8 stay within
    // each 16-lane half. ds_swizzle SWAPX1/2/4/8 patterns.
    #pragma unroll
    for (int r = 0; r < 8; ++r) {
        SWZ_ADD(acc[r], 0x041F);   // xor 1
        SWZ_ADD(acc[r], 0x081F);   // xor 2
        SWZ_ADD(acc[r], 0x101F);   // xor 4
        SWZ_ADD(acc[r], 0x201F);   // xor 8
    }
    if (lane == 0) {
        #pragma unroll
        for (int r = 0; r < 8; ++r) out[b0 + r] = acc[r] + cval;
    } else if (lane == 16) {
        #pragma unroll
        for (int r = 0; r < 8; ++r) out[b0 + 8 + r] = acc[r] + cval;
    }
}

extern "C" void kernel_launch(void* const* d_in, const int* in_sizes, int n_in,
                              void* d_out, int out_size, void* d_ws, size_t ws_size,
                              hipStream_t stream) {
    const float* input = (const float*)d_in[0];
    const float* eq    = (const float*)d_in[1];
    const float* qx0   = (const float*)d_in[2];
    const float* qx1   = (const float*)d_in[3];
    const float* qx2   = (const float*)d_in[4];
    const float* Wx1   = (const float*)d_in[5];
    const float* bx1   = (const float*)d_in[6];
    const float* Wx2   = (const float*)d_in[7];
    const float* bx2   = (const float*)d_in[8];
    const float* Wq01  = (const float*)d_in[9];
    const float* bq01  = (const float*)d_in[10];
    const float* Wq02  = (const float*)d_in[11];
    const float* bq02  = (const float*)d_in[12];
    const float* Wq11  = (const float*)d_in[13];
    const float* bq11  = (const float*)d_in[14];
    const float* Wq12  = (const float*)d_in[15];
    const float* bq12  = (const float*)d_in[16];
    const float* Wq21  = (const float*)d_in[17];
    const float* bq21  = (const float*)d_in[18];
    const float* Wq22  = (const float*)d_in[19];
    const float* bq22  = (const float*)d_in[20];

    float* ws  = (float*)d_ws;
    float* out = (float*)d_out;

    kZero<<<1, 512, 0, stream>>>(ws);
    kBranch<<<96, 128, 0, stream>>>(eq, qx0, qx1, qx2,
                                    Wq01, bq01, Wq11, bq11, Wq21, bq21, ws);
    kCombine<<<1, 512, 0, stream>>>(Wq02, bq02, Wq12, bq12, Wq22, bq22,
                                    Wx2, bx2, ws);
    kMain<<<(NB / 64), 128, 0, stream>>>(input, Wx1, bx1, ws, out);
}